// YOLOLoss_35845797053068
// MI455X (gfx1250) — compile-verified
//
#include <hip/hip_runtime.h>
#include <cstdint>

// ---------------------------------------------------------------------------
// YOLO objectness loss (3 levels) for MI455X / gfx1250.
// Memory-bound & tiny (~8.9 MB touched): 4 small kernels, wave32 everywhere.
// Cross-lane reductions use V_WMMA_F32_16X16X4_F32 with a ones matrix
// (exact f32, deterministic) so the hot path exercises the CDNA5 WMMA unit.
// ---------------------------------------------------------------------------

typedef float v2f __attribute__((ext_vector_type(2)));
typedef float v8f __attribute__((ext_vector_type(8)));

#define BATCH   64
#define NCH     11
#define OBJCH   4
#define W0      160
#define W1      80
#define W2      40
#define P0      (W0*W0)          /* 25600 */
#define P1      (W1*W1)          /* 6400  */
#define P2      (W2*W2)          /* 1600  */
#define BASE0   0
#define BASE1   (BATCH*P0)       /* 1638400 */
#define BASE2   (BASE1 + BATCH*P1) /* 2048000 */
#define ETOT    (BASE2 + BATCH*P2) /* 2150400 */
#define MASK_BYTES (ETOT/8)      /* 268800, 16B- and 256B-aligned size */
#define MAIN_BLOCKS 1024
#define MAIN_THREADS 256

// Exact 32-lane f32 sum via 5x v_wmma_f32_16x16x4_f32.
// Stage 1: A = per-lane partials (A[m][0]=p[m], A[m][2]=p[m+16]), B = ones
//          -> D1[m][n] = p[m] + p[m+16] (independent of n).
// Stage 2: A = ones; B built from D1 VGPR pairs (B[k][n] taps q[2j],q[2j+8],
//          q[2j+1],q[2j+9]); 4 accumulating WMMAs -> every element = total.
__device__ __forceinline__ float wave_reduce32_wmma(float v) {
  v2f a;   a[0] = v;    a[1] = 0.0f;
  v2f one; one[0] = 1.0f; one[1] = 1.0f;
  v8f z = {};
  v8f d1 = __builtin_amdgcn_wmma_f32_16x16x4_f32(false, a, false, one,
                                                 (short)0, z, false, false);
  v8f acc = {};
  v2f b;
  b[0] = d1[0]; b[1] = d1[1];
  acc = __builtin_amdgcn_wmma_f32_16x16x4_f32(false, one, false, b,
                                              (short)0, acc, false, false);
  b[0] = d1[2]; b[1] = d1[3];
  acc = __builtin_amdgcn_wmma_f32_16x16x4_f32(false, one, false, b,
                                              (short)0, acc, false, false);
  b[0] = d1[4]; b[1] = d1[5];
  acc = __builtin_amdgcn_wmma_f32_16x16x4_f32(false, one, false, b,
                                              (short)0, acc, false, false);
  b[0] = d1[6]; b[1] = d1[7];
  acc = __builtin_amdgcn_wmma_f32_16x16x4_f32(false, one, false, b,
                                              (short)0, acc, false, false);
  return acc[0];
}

// ---------------------------------------------------------------------------
// Kernel 1: zero the 1-bit-per-cell hit mask (268800 B = 16800 uint4 stores).
// ---------------------------------------------------------------------------
__global__ void zero_mask_kernel(uint4* __restrict__ m, int n16) {
  int stride = gridDim.x * blockDim.x;
  uint4 z; z.x = 0u; z.y = 0u; z.z = 0u; z.w = 0u;
  for (int i = blockIdx.x * blockDim.x + threadIdx.x; i < n16; i += stride)
    m[i] = z;
}

// ---------------------------------------------------------------------------
// Kernel 2: scatter targets -> hit bits. atomicOr is idempotent, matching the
// reference's .at[...].set(1.0) dedup semantics exactly.
// ---------------------------------------------------------------------------
__device__ __forceinline__ void scatter_level(uint32_t* __restrict__ mask,
                                              int elemBase, int w, int plane,
                                              int b, float fx, float fy) {
  int gx = (int)(fx * (float)w);   // trunc toward zero == astype(int32)
  int gy = (int)(fy * (float)w);   // h == w at every level
  if (gx >= 0 && gx < w && gy >= 0 && gy < w) {
    int e = elemBase + b * plane + gy * w + gx;
    atomicOr(&mask[e >> 5], 1u << (e & 31));
  }
}

__global__ void scatter_targets_kernel(const float* __restrict__ tg, int n,
                                       uint32_t* __restrict__ mask) {
  int t = blockIdx.x * blockDim.x + threadIdx.x;
  if (t >= n) return;
  float bf = tg[t * 6 + 0];
  float fx = tg[t * 6 + 1];
  float fy = tg[t * 6 + 2];
  if (!(bf >= 0.0f && bf < (float)BATCH)) return;   // ref: b < bs (drop OOB)
  int b = (int)bf;
  scatter_level(mask, BASE0, W0, P0, b, fx, fy);
  scatter_level(mask, BASE1, W1, P1, b, fx, fy);
  scatter_level(mask, BASE2, W2, P2, b, fx, fy);
}

// ---------------------------------------------------------------------------
// Kernel 3: fused 3-level BCE partial sums.
//   t=0 cell: min(softplus(x), 100); t=1 cell: min(softplus(-x), 100)
//   (== -max(log1p(-sigmoid),-100) / -max(log(sigmoid),-100))
// Each group = 4 consecutive cells (float4 load of the channel-4 plane).
// ---------------------------------------------------------------------------
template <int PLANE, int BASE>
__device__ __forceinline__ float group_contrib(const float* __restrict__ src,
                                               const uint32_t* __restrict__ mask,
                                               int i) {
  int j   = i - BASE;
  int b   = j / PLANE;               // compile-time-constant divisor
  int pos = j - b * PLANE;
  const float4 x4 =
      *reinterpret_cast<const float4*>(src + (size_t)(b * NCH + OBJCH) * PLANE + pos);
  uint32_t mw = mask[i >> 5];
  int sh = i & 31;
  float xs[4] = {x4.x, x4.y, x4.z, x4.w};
  float s = 0.0f;
#pragma unroll
  for (int k = 0; k < 4; ++k) {
    float x  = xs[k];
    float ax = __builtin_fabsf(x);
    float l  = __logf(1.0f + __expf(-ax));   // log1p(exp(-|x|)), stable
    float sp = fmaxf(x, 0.0f) + l;           // softplus(x)
    float sn = fmaxf(-x, 0.0f) + l;          // softplus(-x)
    float c  = ((mw >> (sh + k)) & 1u) ? sn : sp;
    s += fminf(c, 100.0f);
  }
  return s;
}

__global__ void obj_loss_partial_kernel(const float* __restrict__ p0,
                                        const float* __restrict__ p1,
                                        const float* __restrict__ p2,
                                        const uint32_t* __restrict__ mask,
                                        float* __restrict__ partials) {
  const float inv0 = 1.0f / (float)(BATCH * P0);
  const float inv1 = 1.0f / (float)(BATCH * P1);
  const float inv2 = 1.0f / (float)(BATCH * P2);
  const int nGroups = ETOT / 4;
  const int stride  = gridDim.x * blockDim.x;
  float acc = 0.0f;
  for (int g = blockIdx.x * blockDim.x + threadIdx.x; g < nGroups; g += stride) {
    int i = g * 4;
    if (i < BASE1)      acc += group_contrib<P0, BASE0>(p0, mask, i) * inv0;
    else if (i < BASE2) acc += group_contrib<P1, BASE1>(p1, mask, i) * inv1;
    else                acc += group_contrib<P2, BASE2>(p2, mask, i) * inv2;
  }
  // Wave reduction (WMMA), then per-block sum via LDS. Deterministic.
  float wsum = wave_reduce32_wmma(acc);
  __shared__ float lds[MAIN_THREADS / 32];
  int lane = threadIdx.x & 31, wid = threadIdx.x >> 5;
  if (lane == 0) lds[wid] = wsum;
  __syncthreads();
  if (threadIdx.x == 0) {
    float s = 0.0f;
#pragma unroll
    for (int k = 0; k < MAIN_THREADS / 32; ++k) s += lds[k];
    partials[blockIdx.x] = s;
  }
}

// ---------------------------------------------------------------------------
// Kernel 4: one wave folds the fixed block-partial array into the scalar.
// ---------------------------------------------------------------------------
__global__ void final_reduce_kernel(const float* __restrict__ partials, int n,
                                    float* __restrict__ out) {
  float a = 0.0f;
  for (int i = threadIdx.x; i < n; i += 32) a += partials[i];
  float tot = wave_reduce32_wmma(a);
  if (threadIdx.x == 0) out[0] = tot;
}

// ---------------------------------------------------------------------------
extern "C" void kernel_launch(void* const* d_in, const int* in_sizes, int n_in,
                              void* d_out, int out_size, void* d_ws, size_t ws_size,
                              hipStream_t stream) {
  const float* p0 = (const float*)d_in[0];
  const float* p1 = (const float*)d_in[1];
  const float* p2 = (const float*)d_in[2];
  const float* tg = (const float*)d_in[3];
  int nTgt = in_sizes[3] / 6;

  uint32_t* mask     = (uint32_t*)d_ws;                       // 268800 B
  float*    partials = (float*)((char*)d_ws + MASK_BYTES);    // MAIN_BLOCKS f32

  const int n16 = MASK_BYTES / 16;
  zero_mask_kernel<<<(n16 + 255) / 256, 256, 0, stream>>>((uint4*)d_ws, n16);
  scatter_targets_kernel<<<(nTgt + 255) / 256, 256, 0, stream>>>(tg, nTgt, mask);
  obj_loss_partial_kernel<<<MAIN_BLOCKS, MAIN_THREADS, 0, stream>>>(p0, p1, p2,
                                                                    mask, partials);
  final_reduce_kernel<<<1, 32, 0, stream>>>(partials, MAIN_BLOCKS, (float*)d_out);
}